// BioConvolution_79319456022718
// MI455X (gfx1250) — compile-verified
//
#include <hip/hip_runtime.h>

typedef __attribute__((ext_vector_type(2))) float v2f;
typedef __attribute__((ext_vector_type(4))) float v4f;
typedef __attribute__((ext_vector_type(8))) float v8f;

#define PATCH_K 512          // FH*FW*FIN = 4*4*32
#define KC      256          // K chunk staged in LDS
#define NB      32           // FOUT
#define A_PITCH 260          // 256 + 4 pad floats -> 1040B row, 4-bank lane rotation
#define B_PITCH 72           // 64 + 8 pad floats per k-pair row (288B), 8-bank rotation
#define OUT_M_STRIDE (1024 * 32)   // out[(m*1024 + p)*32 + n]

__launch_bounds__(128)
__global__ void lc_gemm_wmma(const float* __restrict__ X,
                             const float* __restrict__ F,
                             const float* __restrict__ bias,
                             float* __restrict__ O)
{
    // A[m][k] padded; B stored K-pair interleaved: Bs[k2*B_PITCH + n*2 + q] = B[2*k2+q][n]
    __shared__ float As[32 * A_PITCH];       // ~32.5 KB
    __shared__ float Bs[(KC / 2) * B_PITCH]; // 36 KB

    const int p    = blockIdx.x;            // patch id 0..1023
    const int pr   = p >> 5;
    const int pc   = p & 31;
    const int tid  = threadIdx.x;           // 0..127
    const int lane = tid & 31;
    const int wave = tid >> 5;              // 0..3 -> 2x2 C tiles
    const int mbase = (wave >> 1) << 4;     // 0 or 16 (batch tile)
    const int nbase = (wave & 1) << 4;      // 0 or 16 (fout tile)
    const int half  = lane >> 4;            // K sub-pair select per WMMA layout
    const int l16   = lane & 15;

    const float* __restrict__ Fp = F + (size_t)p * (PATCH_K * NB);

    // Per-wave constant LDS fragment bases; all k-stepping via immediate offsets
    const float* __restrict__ Afrag = &As[(mbase + l16) * A_PITCH + 2 * half];
    const float* __restrict__ Bfrag = &Bs[half * B_PITCH + (nbase + l16) * 2];

    v8f acc = {};

    for (int kc0 = 0; kc0 < PATCH_K; kc0 += KC) {
        // ---- stage A chunk: 32 m-rows x 256 k (2048 float4, 16 per thread) ----
        #pragma unroll
        for (int it = 0; it < 16; ++it) {
            int t  = it * 128 + tid;          // float4 id
            int f  = t << 2;                  // flat float id in chunk
            int m  = f >> 8;                  // 0..31
            int kl = f & 255;                 // k within chunk
            int i  = (kc0 >> 7) + (kl >> 7);  // patch pixel row 0..3
            int r  = kl & 127;                // j*FIN + c, 128-float contiguous run
            const float* g =
                X + ((((size_t)m * 128) + (size_t)(pr * 4 + i)) * 128 + pc * 4) * 32 + r;
            *(v4f*)&As[m * A_PITCH + kl] = *(const v4f*)g;
        }
        // ---- stage B chunk with K-pair interleave ----
        // task: (k2, ngroup-of-4); 128*8 = 1024 tasks, 8 per thread
        #pragma unroll
        for (int it = 0; it < 8; ++it) {
            int t  = it * 128 + tid;
            int ng = t & 7;                   // n group (4 columns)
            int k2 = t >> 3;                  // k pair 0..127
            const float* gk = Fp + (kc0 + 2 * k2) * NB + 4 * ng;
            v4f g0 = *(const v4f*)(gk);       // B[2k2  ][n..n+3]
            v4f g1 = *(const v4f*)(gk + NB);  // B[2k2+1][n..n+3]
            float* dst = &Bs[k2 * B_PITCH + 8 * ng];
            *(v4f*)(dst)     = (v4f){g0.x, g1.x, g0.y, g1.y};
            *(v4f*)(dst + 4) = (v4f){g0.z, g1.z, g0.w, g1.w};
        }
        __syncthreads();

        // ---- compute: 64 fully-unrolled V_WMMA_F32_16X16X4_F32 steps ----
        // Linear induction -> every LDS address is base + compile-time immediate:
        //   A: ds_load_2addr_b64, pair offsets 2*ks / 2*ks+2 (8B units, max 126)
        //   B: ds_load_b64 offset: ks*576 bytes (max 36288 < 64KB)
        #pragma unroll
        for (int ks = 0; ks < KC / 4; ++ks) {
            v2f a = *(const v2f*)(Afrag + ks * 4);
            v2f b = *(const v2f*)(Bfrag + ks * (2 * B_PITCH));
            acc = __builtin_amdgcn_wmma_f32_16x16x4_f32(
                      /*neg_a=*/false, a, /*neg_b=*/false, b,
                      /*c_mod=*/(short)0, acc, /*reuse_a=*/false, /*reuse_b=*/false);
        }
        __syncthreads();
    }

    // ---- epilogue: bias + ReLU; C layout: VGPR r -> M = mbase + r + 8*half ----
    const float bv = bias[nbase + l16];
    float* __restrict__ Op = O + (size_t)p * 32 + nbase + l16;
    #pragma unroll
    for (int r = 0; r < 8; ++r) {
        int   m = mbase + r + 8 * half;
        float v = acc[r] + bv;
        v = v > 0.0f ? v : 0.0f;
        Op[(size_t)m * OUT_M_STRIDE] = v;
    }
}

extern "C" void kernel_launch(void* const* d_in, const int* in_sizes, int n_in,
                              void* d_out, int out_size, void* d_ws, size_t ws_size,
                              hipStream_t stream) {
    const float* X    = (const float*)d_in[0];
    const float* F    = (const float*)d_in[1];
    const float* bias = (const float*)d_in[2];
    float*       O    = (float*)d_out;

    dim3 grid(1024);   // one workgroup per patch-GEMM (M=32, N=32, K=512)
    dim3 block(128);   // 4 wave32 -> 2x2 WMMA C tiles
    hipLaunchKernelGGL(lc_gemm_wmma, grid, block, 0, stream, X, F, bias, O);
}